// HCMaskLayer_8856222564441
// MI455X (gfx1250) — compile-verified
//
#include <hip/hip_runtime.h>
#include <hip/hip_bf16.h>

// ---------------------------------------------------------------------------
// HCMaskLayer region means on MI455X (gfx1250, wave32).
//
// Memory-bound region reduction (1 add / 4 bytes @ 23.3 TB/s HBM). Inner
// reduction is D = ones(16x4) x B(4x16) + C via V_WMMA_F32_16X16X4_F32.
// Column->channel mapping is bookkeeping only, so each lane loads float4 =
// 4 consecutive channels: one global_load_b128 moves 512B and 2 loads feed
// 4 WMMAs covering all 64 channels x 4 y-rows. Y-stripe id is forced into an
// SGPR via readfirstlane so all loop control is scalar (s_cmp/s_cbranch, no
// exec masking); unroll-4 lets the scheduler clause 8 loads ahead of 16 WMMAs.
// ---------------------------------------------------------------------------

#define NDIM 2048
#define CH   64

typedef __attribute__((ext_vector_type(2))) float v2f;
typedef __attribute__((ext_vector_type(8))) float v8f;

struct Acc4 { v8f a0, a1, a2, a3; };

// b.x = B[k=2*khalf, n] (y even), b.y = B[k=2*khalf+1, n] (y odd).
// A == ones -> D[m,n] = sum_k B[k,n] + C[m,n]; all 16 D rows identical.
__device__ __forceinline__ void wmma_accum(Acc4& acc, const v2f& ones,
                                           const float4& L0, const float4& L1) {
    v2f b;
    b.x = L0.x; b.y = L1.x;
    acc.a0 = __builtin_amdgcn_wmma_f32_16x16x4_f32(false, ones, false, b,
                                                   (short)0, acc.a0, false, false);
    b.x = L0.y; b.y = L1.y;
    acc.a1 = __builtin_amdgcn_wmma_f32_16x16x4_f32(false, ones, false, b,
                                                   (short)0, acc.a1, false, false);
    b.x = L0.z; b.y = L1.z;
    acc.a2 = __builtin_amdgcn_wmma_f32_16x16x4_f32(false, ones, false, b,
                                                   (short)0, acc.a2, false, false);
    b.x = L0.w; b.y = L1.w;
    acc.a3 = __builtin_amdgcn_wmma_f32_16x16x4_f32(false, ones, false, b,
                                                   (short)0, acc.a3, false, false);
}

// One block per (region r, row-slice s). 256 threads = 8 waves; each wave owns
// a y-stripe of 4 rows (step 32) and covers all 64 channels via 4 accumulators.
// Cross-wave combine is a fixed-order LDS reduction (bitwise deterministic).
__global__ __launch_bounds__(256) void hc_region_sums(
    const float* __restrict__ img,
    const int* __restrict__ rx0, const int* __restrict__ rx1,
    const int* __restrict__ ry0, const int* __restrict__ ry1,
    float* __restrict__ part, int splits)
{
    __shared__ float lsum[8][CH];

    const int r = blockIdx.x / splits;
    const int s = blockIdx.x - r * splits;

    const int X0 = rx0[r], X1 = rx1[r];
    const int Y0 = ry0[r], Y1 = ry1[r];

    const int tid = threadIdx.x;
    // Wave id is uniform across the 32 lanes; readfirstlane makes it an SGPR
    // so the y-loop below gets scalar control flow (no exec masking).
    const int wave = __builtin_amdgcn_readfirstlane(tid >> 5);   // 0..7
    const int lane = tid & 31;
    const int n    = lane & 15;          // B column; channels 4n..4n+3
    const int yoff = (lane >> 4) * 2;    // khalf: lanes 16-31 take y+2, y+3

    const float4* __restrict__ img4 = reinterpret_cast<const float4*>(img);

    v2f ones; ones.x = 1.0f; ones.y = 1.0f;
    Acc4 acc; acc.a0 = (v8f){}; acc.a1 = (v8f){}; acc.a2 = (v8f){}; acc.a3 = (v8f){};

    const int ystart = Y0 + wave * 4;    // scalar

    for (int x = X0 + s; x < X1; x += splits) {
        // float4 index of (x, y, 4n): (x*NDIM + y)*16 + n
        const float4* __restrict__ p =
            img4 + ((size_t)x * NDIM + (size_t)(ystart + yoff)) * 16 + (size_t)n;

        int yb = ystart;
        // ---- main loop: full 4-row tiles, guard-free, scalar-controlled ----
        #pragma unroll 4
        for (; yb + 4 <= Y1; yb += 32) {
            const float4 L0 = p[0];      // y = yb + yoff
            const float4 L1 = p[16];     // y = yb + yoff + 1  (+256 B)
            p += 32 * 16;                // next stripe tile (+8 KiB)
            wmma_accum(acc, ones, L0, L1);
        }
        // ---- tail: branchless clamped loads + per-lane zero masks ----
        if (yb < Y1) {                   // scalar condition
            const int yA = yb + yoff, yB = yA + 1;
            const int yAc = (yA < Y1) ? yA : (Y1 - 1);
            const int yBc = (yB < Y1) ? yB : (Y1 - 1);
            const size_t rowi = ((size_t)x * NDIM) * 16 + (size_t)n;
            float4 vA = img4[rowi + (size_t)yAc * 16];
            float4 vB = img4[rowi + (size_t)yBc * 16];
            const float mA = (yA < Y1) ? 1.0f : 0.0f;
            const float mB = (yB < Y1) ? 1.0f : 0.0f;
            vA.x *= mA; vA.y *= mA; vA.z *= mA; vA.w *= mA;
            vB.x *= mB; vB.y *= mB; vB.z *= mB; vB.w *= mB;
            wmma_accum(acc, ones, vA, vB);
        }
    }

    // D row 0 lives in accumulator VGPR0, lanes 0-15 (lanes 16-31 duplicate).
    if (lane < 16) {
        lsum[wave][4 * n + 0] = acc.a0[0];
        lsum[wave][4 * n + 1] = acc.a1[0];
        lsum[wave][4 * n + 2] = acc.a2[0];
        lsum[wave][4 * n + 3] = acc.a3[0];
    }
    __syncthreads();

    if (tid < CH) {
        float ssum = 0.0f;
        #pragma unroll
        for (int w = 0; w < 8; ++w) ssum += lsum[w][tid];   // fixed order
        part[((size_t)r * splits + s) * CH + tid] = ssum;
    }
}

// One block per region, CH threads: fixed-order sum over `splits` partials,
// divide by pixel count, 0 for empty regions. Safe when part == out (splits 1).
__global__ __launch_bounds__(CH) void hc_finalize(
    const float* __restrict__ part,
    const int* __restrict__ rx0, const int* __restrict__ rx1,
    const int* __restrict__ ry0, const int* __restrict__ ry1,
    float* __restrict__ out, int splits)
{
    const int r = blockIdx.x;
    const int c = threadIdx.x;
    const long long cnt =
        (long long)(rx1[r] - rx0[r]) * (long long)(ry1[r] - ry0[r]);
    float ssum = 0.0f;
    for (int i = 0; i < splits; ++i) {
        ssum += part[((size_t)r * splits + i) * CH + c];
    }
    out[(size_t)r * CH + c] = (cnt > 0) ? (ssum / (float)cnt) : 0.0f;
}

extern "C" void kernel_launch(void* const* d_in, const int* in_sizes, int n_in,
                              void* d_out, int out_size, void* d_ws, size_t ws_size,
                              hipStream_t stream) {
    const float* img = (const float*)d_in[0];
    const int*   x0  = (const int*)d_in[1];
    const int*   x1  = (const int*)d_in[2];
    const int*   y0  = (const int*)d_in[3];
    const int*   y1  = (const int*)d_in[4];

    const int R = in_sizes[1];                 // 64 regions
    float* outp = (float*)d_out;

    // Row-slices per region, bounded by workspace capacity (R*CH floats per
    // slice). If ws can't hold even one slice, accumulate straight into d_out
    // (splits = 1); finalize then normalizes in place.
    const size_t per_split_bytes = (size_t)R * CH * sizeof(float);
    int    splits = 1;
    float* part   = outp;
    if (ws_size >= per_split_bytes) {
        splits = (int)(ws_size / per_split_bytes);
        if (splits > 32) splits = 32;
        part = (float*)d_ws;
    }

    dim3 grid_main((unsigned)(R * splits), 1, 1);
    dim3 blk_main(256, 1, 1);
    hc_region_sums<<<grid_main, blk_main, 0, stream>>>(img, x0, x1, y0, y1,
                                                       part, splits);

    dim3 grid_fin((unsigned)R, 1, 1);
    dim3 blk_fin(CH, 1, 1);
    hc_finalize<<<grid_fin, blk_fin, 0, stream>>>(part, x0, x1, y0, y1,
                                                  outp, splits);
}